// XmmLayer_64570538328242
// MI455X (gfx1250) — compile-verified
//
#include <hip/hip_runtime.h>
#include <hip/hip_bf16.h>
#include <stdint.h>

#ifndef __has_builtin
#define __has_builtin(x) 0
#endif

namespace {
constexpr int kB = 2048;   // batch
constexpr int kI = 256;    // in_features
constexpr int kO = 256;    // out_features
constexpr float kEps = 1e-5f;

constexpr int BT  = 64;    // batch tile per workgroup
constexpr int OT  = 64;    // output tile per workgroup
constexpr int IC  = 32;    // i-chunk staged in LDS
constexpr int NC  = kI / IC;  // number of i-chunks
constexpr int PAD = 33;    // LDS row stride (words): gcd(33,64)=1 -> conflict-free
}  // namespace

#if defined(__HIP_DEVICE_COMPILE__)
#if __has_builtin(__builtin_amdgcn_global_load_async_to_lds_b32)
#define XMM_ASYNC 1
#else
#define XMM_ASYNC 0
#endif
#if __has_builtin(__builtin_amdgcn_wmma_f32_16x16x4_f32)
#define XMM_WMMA 1
#else
#define XMM_WMMA 0
#endif
#else
#define XMM_ASYNC 0
#define XMM_WMMA 0
#endif

typedef float v2f __attribute__((ext_vector_type(2)));
typedef float v8f __attribute__((ext_vector_type(8)));

// ---- CDNA5 async global->LDS copy (ASYNCcnt path), with safe fallback ----
__device__ __forceinline__ void ld_async_b32(const float* g, float* l) {
#if XMM_ASYNC
  __builtin_amdgcn_global_load_async_to_lds_b32(
      (__attribute__((address_space(1))) int*)g,
      (__attribute__((address_space(3))) int*)l,
      /*offset=*/0, /*cpol=*/0);
#else
  *l = *g;
#endif
}

__device__ __forceinline__ void wait_async_zero() {
#if XMM_ASYNC
#if __has_builtin(__builtin_amdgcn_s_wait_asynccnt)
  __builtin_amdgcn_s_wait_asynccnt(0);
#else
  asm volatile("s_wait_asynccnt 0" ::: "memory");
#endif
#endif
}

// Issue the async fill for one i-chunk into one LDS buffer set.
__device__ __forceinline__ void stage_chunk(
    const float* __restrict__ x, const float* __restrict__ scale,
    const float* __restrict__ bias, const float* __restrict__ weight,
    float* xs, float* ss, float* cs, float* wgt,
    int tid, int b0, int o0, int i0) {
#pragma unroll
  for (int k = 0; k < 8; ++k) {          // xs: 64x32 = 2048 words / 256 thr
    int w = tid + k * 256;
    int r = w >> 5, c = w & 31;
    ld_async_b32(&x[(b0 + r) * kI + i0 + c], &xs[r * PAD + c]);
  }
#pragma unroll
  for (int k = 0; k < 8; ++k) {          // s/c/w: 64x32 = 2048 words each
    int w = tid + k * 256;
    int r = w >> 5, c = w & 31;
    int gofs = (o0 + r) * kI + i0 + c;
    ld_async_b32(&scale[gofs],  &ss[r * PAD + c]);
    ld_async_b32(&bias[gofs],   &cs[r * PAD + c]);
    ld_async_b32(&weight[gofs], &wgt[r * PAD + c]);
  }
}

// =====================================================================
// Phase 1: y[b,o] = sum_i w[o,i] * (g-1) * exp(-0.5 g^2),
//          g = s[o,i]*(x[b,i] + c[o,i])
// Tile: 64 batches x 64 outputs per 256-thread workgroup (8 waves).
// Each wave owns 8 consecutive outputs (wave-uniform -> LDS broadcast
// reads for s/c/w). Each lane owns 2 batches -> 16 accumulators.
// Double-buffered async staging: chunk c+1 streams into the alternate
// LDS buffer while the VALU/trans pipes chew on chunk c.
// =====================================================================
__global__ __launch_bounds__(256) void xmm_phase1(
    const float* __restrict__ x, const float* __restrict__ scale,
    const float* __restrict__ bias, const float* __restrict__ weight,
    float* __restrict__ y) {
  __shared__ float xs[2][BT * PAD];
  __shared__ float ss[2][OT * PAD];
  __shared__ float cs[2][OT * PAD];
  __shared__ float wgt[2][OT * PAD];

  const int tid  = threadIdx.x;
  const int lane = tid & 31;
  const int wave = tid >> 5;              // 0..7
  const int b0   = blockIdx.x * BT;
  const int o0   = blockIdx.y * OT;
  const int oW   = wave * 8;              // wave-uniform output sub-tile

  float acc0[8], acc1[8];
#pragma unroll
  for (int j = 0; j < 8; ++j) { acc0[j] = 0.f; acc1[j] = 0.f; }

  // prologue: fill buffer 0 with chunk 0
  stage_chunk(x, scale, bias, weight, xs[0], ss[0], cs[0], wgt[0],
              tid, b0, o0, 0);

  for (int ci = 0; ci < NC; ++ci) {
    // My async issues (current buffer, and possibly next) are complete;
    // barrier publishes LDS and guarantees everyone is done reading the
    // buffer we are about to refill.
    wait_async_zero();
    __syncthreads();

    const int cur = ci & 1;
    if (ci + 1 < NC) {
      stage_chunk(x, scale, bias, weight, xs[cur ^ 1], ss[cur ^ 1],
                  cs[cur ^ 1], wgt[cur ^ 1], tid, b0, o0, (ci + 1) * IC);
    }

    const float* xsb = xs[cur];
    const float* ssb = ss[cur];
    const float* csb = cs[cur];
    const float* wgb = wgt[cur];

    // ---- fused basis + weighted reduction (overlaps with async fill) ----
    for (int ii = 0; ii < IC; ++ii) {
      const float xv0 = xsb[lane * PAD + ii];
      const float xv1 = xsb[(lane + 32) * PAD + ii];
#pragma unroll
      for (int j = 0; j < 8; ++j) {
        const int oi = (oW + j) * PAD + ii;
        const float s = ssb[oi];            // wave-uniform -> broadcast
        const float c = csb[oi];
        const float w = wgb[oi];
        const float g0 = s * (xv0 + c);
        const float g1 = s * (xv1 + c);
        acc0[j] = fmaf(w * (g0 - 1.f), __expf(-0.5f * g0 * g0), acc0[j]);
        acc1[j] = fmaf(w * (g1 - 1.f), __expf(-0.5f * g1 * g1), acc1[j]);
      }
    }
  }

  // coalesced-per-lane 32B stores (o0+oW is a multiple of 8)
  float4* p0 = (float4*)&y[(b0 + lane) * kO + o0 + oW];
  float4* p1 = (float4*)&y[(b0 + lane + 32) * kO + o0 + oW];
  p0[0] = make_float4(acc0[0], acc0[1], acc0[2], acc0[3]);
  p0[1] = make_float4(acc0[4], acc0[5], acc0[6], acc0[7]);
  p1[0] = make_float4(acc1[0], acc1[1], acc1[2], acc1[3]);
  p1[1] = make_float4(acc1[4], acc1[5], acc1[6], acc1[7]);
}

// =====================================================================
// Phase 2: per-column batch statistics via f32 WMMA.
// D = ones(16x4) x Y_chunk(4x16) accumulates column sums in every row
// of the 16x16 accumulator; same with Y^2 for sumsq. Full f32 precision.
// One block per 16 output columns, 4 waves split the batch.
// =====================================================================
__global__ __launch_bounds__(128) void bn_stats(
    const float* __restrict__ y, const float* __restrict__ gamma,
    const float* __restrict__ beta, float* __restrict__ stats) {
  const int o0   = blockIdx.x * 16;
  const int tid  = threadIdx.x;
  const int lane = tid & 31;
  const int wave = tid >> 5;  // 0..3

  __shared__ float psum[4][16];
  __shared__ float psq[4][16];

#if XMM_WMMA
  v8f accS = {}, accQ = {};
  const v2f ones = {1.f, 1.f};
  const int col   = lane & 15;
  const int rbase = (lane >> 4) << 1;  // 0 or 2 (B-matrix f32 layout)
  for (int bb = wave * 4; bb < kB; bb += 16) {
    v2f bv;
    bv.x = y[(bb + rbase) * kO + o0 + col];
    bv.y = y[(bb + rbase + 1) * kO + o0 + col];
    v2f bq = bv * bv;
    accS = __builtin_amdgcn_wmma_f32_16x16x4_f32(
        false, ones, false, bv, (short)0, accS, false, false);
    accQ = __builtin_amdgcn_wmma_f32_16x16x4_f32(
        false, ones, false, bq, (short)0, accQ, false, false);
  }
  if (lane < 16) {  // row m=0 of D holds the column sums
    psum[wave][lane] = accS[0];
    psq[wave][lane]  = accQ[0];
  }
#else
  float s = 0.f, q = 0.f;
  if (lane < 16) {
    for (int b = wave; b < kB; b += 4) {
      float v = y[b * kO + o0 + lane];
      s += v;
      q += v * v;
    }
    psum[wave][lane] = s;
    psq[wave][lane]  = q;
  }
#endif
  __syncthreads();

  if (tid < 16) {
    float s = psum[0][tid] + psum[1][tid] + psum[2][tid] + psum[3][tid];
    float q = psq[0][tid] + psq[1][tid] + psq[2][tid] + psq[3][tid];
    const float inv = 1.f / (float)kB;
    float mean = s * inv;
    float var  = q * inv - mean * mean;  // biased, matches jnp.var
    const int o = o0 + tid;
    float scl = gamma[o] * rsqrtf(var + kEps);
    stats[2 * o]     = scl;
    stats[2 * o + 1] = beta[o] - mean * scl;
  }
}

// =====================================================================
// Phase 3: out = y * scl[o] + sft[o], vectorized float4.
// =====================================================================
__global__ __launch_bounds__(256) void bn_apply(
    const float* __restrict__ y, const float* __restrict__ stats,
    float* __restrict__ out) {
  const int idx = (blockIdx.x * 256 + threadIdx.x) * 4;  // 4 consecutive o
  const int o   = idx & (kO - 1);
  float4 v  = *(const float4*)&y[idx];
  float4 s0 = *(const float4*)&stats[2 * o];        // (scl,sft) x 2
  float4 s1 = *(const float4*)&stats[2 * o + 4];    // (scl,sft) x 2
  float4 r;
  r.x = fmaf(v.x, s0.x, s0.y);
  r.y = fmaf(v.y, s0.z, s0.w);
  r.z = fmaf(v.z, s1.x, s1.y);
  r.w = fmaf(v.w, s1.z, s1.w);
  *(float4*)&out[idx] = r;
}

extern "C" void kernel_launch(void* const* d_in, const int* in_sizes, int n_in,
                              void* d_out, int out_size, void* d_ws,
                              size_t ws_size, hipStream_t stream) {
  (void)in_sizes; (void)n_in; (void)out_size; (void)ws_size;
  const float* x      = (const float*)d_in[0];
  const float* scale  = (const float*)d_in[1];
  const float* bias   = (const float*)d_in[2];
  const float* weight = (const float*)d_in[3];
  const float* gamma  = (const float*)d_in[4];
  const float* beta   = (const float*)d_in[5];
  float* out   = (float*)d_out;
  float* y     = (float*)d_ws;            // 2048*256 f32 = 2 MB
  float* stats = y + kB * kO;             // 512 f32 (scl,sft per column)

  dim3 g1(kB / BT, kO / OT);              // 32 x 4
  xmm_phase1<<<g1, 256, 0, stream>>>(x, scale, bias, weight, y);
  bn_stats<<<kO / 16, 128, 0, stream>>>(y, gamma, beta, stats);
  bn_apply<<<(kB * kO) / (256 * 4), 256, 0, stream>>>(y, stats, out);
}